// User_Graph_sample_9921374454292
// MI455X (gfx1250) — compile-verified
//
#include <hip/hip_runtime.h>
#include <hip/hip_bf16.h>

typedef __attribute__((ext_vector_type(2))) float v2f;
typedef __attribute__((ext_vector_type(8))) float v8f;

#define DIM 64
#define KNB 32
#define WAVES_PER_BLOCK 8

// One wave (32 lanes) per user.
// D(16x16) = A(16x4) x B(4x16) + C  via V_WMMA_F32_16X16X4_F32 (pure f32 path).
//   A[d,k] = features[g[n, kbase+k], tile*16 + d]   (d = M dim, k = K dim)
//   B[k,j] = w[n, kbase+k]  for all columns j        (broadcast -> redundant N)
//   D[d,j] = out[n, tile*16 + d]  (all 16 columns identical)
// ISA layouts (CDNA5 ISA):
//   A 16x4 f32 : lanes 0-15 carry M=0..15; VGPR0={K0 | K2}, VGPR1={K1 | K3}
//   B 4x16 f32 : lanes 0-15 rows K0/K1, lanes 16-31 rows K2/K3 (per VGPR0/1)
//   D 16x16 f32: VGPR r = M=r (lanes 0-15, N=lane) / M=r+8 (lanes 16-31)
__global__ __launch_bounds__(256) void user_graph_wsum_wmma(
    const float* __restrict__ features,
    const int*   __restrict__ graph,   // neighbor indices, 32 per user
    const float* __restrict__ wmat,    // weights, 32 per user
    float*       __restrict__ out,
    int num_user)
{
    const int lane = threadIdx.x & 31;
    const int wave = threadIdx.x >> 5;
    const int user = blockIdx.x * WAVES_PER_BLOCK + wave;
    if (user >= num_user) return;  // wave-uniform: EXEC stays all-1s for WMMA

    // Each lane owns one of the 32 neighbors (index + weight); redistributed
    // via wave32 shuffles below -- indices/weights touch memory exactly once.
    const int   gidx = graph[(size_t)user * KNB + lane];
    const float wv   = wmat [(size_t)user * KNB + lane];

    const int halfSel = (lane >> 4) << 1;   // 0 for lanes 0-15, 2 for 16-31
    const int col     = lane & 15;          // M position this lane serves in A

    v8f acc0 = {}, acc1 = {}, acc2 = {}, acc3 = {};

    #pragma unroll
    for (int ks = 0; ks < KNB / 4; ++ks) {
        const int kbase = ks * 4;

        // B: rows K=kbase..kbase+3, all columns identical (= the weight)
        v2f b;
        b.x = __shfl(wv, kbase + halfSel + 0, 32);
        b.y = __shfl(wv, kbase + halfSel + 1, 32);

        // Neighbor rows feeding this lane's two A slots
        const int r0 = __shfl(gidx, kbase + halfSel + 0, 32);
        const int r1 = __shfl(gidx, kbase + halfSel + 1, 32);
        const float* p0 = features + (size_t)r0 * DIM + col;
        const float* p1 = features + (size_t)r1 * DIM + col;

        v2f a;
        a.x = p0[0];  a.y = p1[0];   // dim tile 0: cols 0..15
        acc0 = __builtin_amdgcn_wmma_f32_16x16x4_f32(false, a, false, b,
                                                     (short)0, acc0, false, false);
        a.x = p0[16]; a.y = p1[16];  // dim tile 1
        acc1 = __builtin_amdgcn_wmma_f32_16x16x4_f32(false, a, false, b,
                                                     (short)0, acc1, false, false);
        a.x = p0[32]; a.y = p1[32];  // dim tile 2
        acc2 = __builtin_amdgcn_wmma_f32_16x16x4_f32(false, a, false, b,
                                                     (short)0, acc2, false, false);
        a.x = p0[48]; a.y = p1[48];  // dim tile 3
        acc3 = __builtin_amdgcn_wmma_f32_16x16x4_f32(false, a, false, b,
                                                     (short)0, acc3, false, false);
    }

    // Column 0 of each D tile carries the answer: lane 0 holds M=0..7,
    // lane 16 holds M=8..15. Contiguous runs -> compiler packs b128 stores.
    float* o = out + (size_t)user * DIM;
    if (lane == 0) {
        #pragma unroll
        for (int r = 0; r < 8; ++r) {
            o[ 0 + r] = acc0[r];
            o[16 + r] = acc1[r];
            o[32 + r] = acc2[r];
            o[48 + r] = acc3[r];
        }
    } else if (lane == 16) {
        #pragma unroll
        for (int r = 0; r < 8; ++r) {
            o[ 8 + r] = acc0[r];
            o[24 + r] = acc1[r];
            o[40 + r] = acc2[r];
            o[56 + r] = acc3[r];
        }
    }
}

extern "C" void kernel_launch(void* const* d_in, const int* in_sizes, int n_in,
                              void* d_out, int out_size, void* d_ws, size_t ws_size,
                              hipStream_t stream) {
    const float* features = (const float*)d_in[0];
    const int*   graph    = (const int*)  d_in[1];  // integer input -> const int*
    const float* wmat     = (const float*)d_in[2];
    float*       out      = (float*)d_out;

    const int num_user = in_sizes[0] / DIM;         // features is [num_user, 64]
    const int blocks   = (num_user + WAVES_PER_BLOCK - 1) / WAVES_PER_BLOCK;

    user_graph_wsum_wmma<<<blocks, WAVES_PER_BLOCK * 32, 0, stream>>>(
        features, graph, wmat, out, num_user);
}